// Kadj_50732153700404
// MI455X (gfx1250) — compile-verified
//
#include <hip/hip_runtime.h>
#include <math.h>

typedef __attribute__((ext_vector_type(2))) float v2f;
typedef __attribute__((ext_vector_type(8))) float v8f;

#define B_   8
#define N_   2048
#define D_   512
#define KSEL 16

// ---------------------------------------------------------------------------
// Kernel 1: squared row norms. One wave32 per row, shuffle reduction.
// grid = B*N/8 blocks of 256 threads (8 waves).
// ---------------------------------------------------------------------------
__global__ __launch_bounds__(256)
void Kadj_norms_kernel(const float* __restrict__ put, float* __restrict__ xnorm) {
    const int lane = threadIdx.x & 31;
    const int row  = blockIdx.x * 8 + (threadIdx.x >> 5);
    const float* p = put + (size_t)row * D_;
    float s = 0.f;
#pragma unroll
    for (int i = 0; i < D_ / 32; ++i) {
        float v = p[lane + i * 32];
        s = fmaf(v, v, s);
    }
#pragma unroll
    for (int off = 16; off > 0; off >>= 1)
        s += __shfl_xor(s, off, 32);
    if (lane == 0) xnorm[row] = s;
}

// ---------------------------------------------------------------------------
// Kernel 2: per (batch, 16-row stripe): WMMA gram -> dist2 stripe in LDS ->
// iterative top-16 per row -> write 0/1 adjacency stripe.
// grid = (N/16, B), block = 256 (8 waves), dyn LDS = 16*512 + 16*2048 floats.
// Each wave processes TWO column tiles at once: two independent WMMA
// accumulator chains sharing one A ds_load (2x XDL ILP, half the LDS reads).
// ---------------------------------------------------------------------------
__global__ __launch_bounds__(256, 1)
void Kadj_main_kernel(const float* __restrict__ put,
                      const float* __restrict__ xnorm,
                      float* __restrict__ out) {
    extern __shared__ float smem[];
    float* As = smem;               // [16][512]  A stripe
    float* Ds = smem + 16 * D_;     // [16][2048] dist2 stripe

    const int b       = blockIdx.y;
    const int rowBase = blockIdx.x * 16;
    const int tid     = threadIdx.x;
    const int wave    = tid >> 5;
    const int lane    = tid & 31;
    const int l16     = lane & 15;
    const int hi      = lane >> 4;        // 0 for lanes 0-15, 1 for 16-31
    const int koff    = hi << 1;          // K sub-offset per A/B VGPR layout

    // ---- stage A stripe (16*512 contiguous floats) into LDS, float4 coalesced
    {
        const float4* gsrc = (const float4*)(put + ((size_t)b * N_ + rowBase) * D_);
        float4* ldst = (float4*)As;
#pragma unroll
        for (int i = 0; i < (16 * D_ / 4) / 256; ++i)   // 8 iterations
            ldst[tid + i * 256] = gsrc[tid + i * 256];
    }
    __syncthreads();

    // ---- per-lane row norms for the 8 M-values this lane's C tile holds
    float rowN[8];
#pragma unroll
    for (int i = 0; i < 8; ++i)
        rowN[i] = xnorm[(size_t)b * N_ + rowBase + (hi << 3) + i];

    // ---- GEMM + distance: wave w owns column tiles {w + 8m}; paired (ct, ct+8)
    for (int ct = wave; ct < N_ / 16; ct += 16) {
        const int c0 = ct * 16 + l16;
        const int c1 = c0 + 128;                       // tile ct+8
        const float* p0   = put + ((size_t)b * N_ + c0) * D_ + koff;
        const float* p1   = put + ((size_t)b * N_ + c1) * D_ + koff;
        const float* aPtr = As + l16 * D_ + koff;

        v8f acc0 = {};
        v8f acc1 = {};
#pragma unroll 4
        for (int k0 = 0; k0 < D_; k0 += 4) {
            v2f a  = *(const v2f*)(aPtr + k0);         // ds_load_b64 (shared)
            v2f b0 = *(const v2f*)(p0 + k0);           // global_load_b64
            v2f b1 = *(const v2f*)(p1 + k0);           // global_load_b64
            acc0 = __builtin_amdgcn_wmma_f32_16x16x4_f32(
                false, a, false, b0, (short)0, acc0, false, false);
            acc1 = __builtin_amdgcn_wmma_f32_16x16x4_f32(
                false, a, false, b1, (short)0, acc1, false, false);
        }

        const float cn0 = xnorm[(size_t)b * N_ + c0];
        const float cn1 = xnorm[(size_t)b * N_ + c1];
#pragma unroll
        for (int i = 0; i < 8; ++i) {
            const int M = (hi << 3) + i;               // C layout: VGPR i, group hi
            Ds[M * N_ + c0] = fmaxf(rowN[i] + cn0 - 2.0f * acc0[i], 0.0f);
            Ds[M * N_ + c1] = fmaxf(rowN[i] + cn1 - 2.0f * acc1[i], 0.0f);
        }
    }
    __syncthreads();

    // ---- top-16 per row; each wave handles rows {wave, wave+8}
    const float INF = __builtin_inff();
    for (int r = wave; r < 16; r += 8) {
        float* rowp = Ds + r * N_;
        const float4* rowv = (const float4*)rowp;      // ds_load_b128 scans
        for (int it = 0; it < KSEL; ++it) {
            float mv = 3.0e38f;
            int   mj = N_;
#pragma unroll 4
            for (int i = 0; i < (N_ / 4) / 32; ++i) {  // 16 vector loads / lane
                const int vi   = lane + i * 32;
                const float4 v = rowv[vi];
                const int base = vi * 4;
                // ascending-index order + strict < keeps lowest index on ties
                if (v.x < mv) { mv = v.x; mj = base;     }
                if (v.y < mv) { mv = v.y; mj = base + 1; }
                if (v.z < mv) { mv = v.z; mj = base + 2; }
                if (v.w < mv) { mv = v.w; mj = base + 3; }
            }
#pragma unroll
            for (int off = 16; off > 0; off >>= 1) {
                float ov = __shfl_xor(mv, off, 32);
                int   oj = __shfl_xor(mj, off, 32);
                if (ov < mv || (ov == mv && oj < mj)) { mv = ov; mj = oj; }
            }
            if (lane == 0) rowp[mj] = INF;             // mark selected
            asm volatile("s_wait_dscnt 0" ::: "memory");
        }
        // emit 0/1 adjacency row (selected entries are +inf markers)
        const size_t obase = ((size_t)b * N_ + rowBase + r) * (size_t)N_;
        float4* outv = (float4*)(out + obase);         // coalesced b128 stores
#pragma unroll 4
        for (int i = 0; i < (N_ / 4) / 32; ++i) {
            const int vi   = lane + i * 32;
            const float4 v = rowv[vi];
            float4 o;
            o.x = (v.x == INF) ? 1.0f : 0.0f;
            o.y = (v.y == INF) ? 1.0f : 0.0f;
            o.z = (v.z == INF) ? 1.0f : 0.0f;
            o.w = (v.w == INF) ? 1.0f : 0.0f;
            outv[vi] = o;
        }
    }
}

// ---------------------------------------------------------------------------
extern "C" void kernel_launch(void* const* d_in, const int* in_sizes, int n_in,
                              void* d_out, int out_size, void* d_ws, size_t ws_size,
                              hipStream_t stream) {
    const float* put = (const float*)d_in[0];
    // d_in[1] is k == 16 (python scalar); compile-time specialized to KSEL.
    float* out   = (float*)d_out;
    float* xnorm = (float*)d_ws;          // B*N floats = 64 KB scratch

    Kadj_norms_kernel<<<(B_ * N_) / 8, 256, 0, stream>>>(put, xnorm);

    const size_t smemBytes = (size_t)(16 * D_ + 16 * N_) * sizeof(float); // 160 KB
    Kadj_main_kernel<<<dim3(N_ / 16, B_), 256, smemBytes, stream>>>(put, xnorm, out);
}